// AfterShockGNN_90159953478465
// MI455X (gfx1250) — compile-verified
//
#include <hip/hip_runtime.h>
#include <hip/hip_bf16.h>

typedef __attribute__((ext_vector_type(2))) float v2f;
typedef __attribute__((ext_vector_type(8))) float v8f;

#define HDIM 64

// ---------------- utility ----------------
__global__ void zero_f32v4(float4* __restrict__ p, long long n4) {
    long long i = (long long)blockIdx.x * blockDim.x + threadIdx.x;
    if (i < n4) p[i] = make_float4(0.f, 0.f, 0.f, 0.f);
}

// deg[dst] += 1 per edge
__global__ void deg_kernel(const int* __restrict__ dst, float* __restrict__ deg, int E) {
    int e = blockIdx.x * blockDim.x + threadIdx.x;
    if (e < E) atomicAdd(&deg[dst[e]], 1.0f);
}

// dinv = rsqrt(deg + 1)   (+1 = self loop)
__global__ void dinv_kernel(float* __restrict__ dinv, int n) {
    int i = blockIdx.x * blockDim.x + threadIdx.x;
    if (i < n) dinv[i] = rsqrtf(dinv[i] + 1.0f);
}

// ---------------- WMMA GEMM: out[N,64] = X[N,K] @ W[K,64] ----------------
// grid.x = N/32 blocks, 64 threads = 2 waves; each wave computes a 16x64 strip
// with 4 accumulators so one A fragment feeds 4 V_WMMA_F32_16X16X4_F32 ops.
// W is staged in LDS k-pair interleaved: sW[(k/2)*128 + n*2 + (k&1)], so each
// lane's B fragment (W[k][n], W[k+1][n]) is one contiguous ds_load_b64.
__global__ void __launch_bounds__(64)
gemm_wmma_f32(const float* __restrict__ X, const float* __restrict__ W,
              float* __restrict__ out, int K) {
    __shared__ float sW[128 * HDIM];          // max K=128 -> 32 KB
    const int tid = threadIdx.x;
    for (int i = tid; i < K * HDIM; i += 64) {
        int k = i >> 6;                        // row of W
        int n = i & 63;                        // col of W
        sW[((k >> 1) << 7) + (n << 1) + (k & 1)] = W[i];
    }
    __syncthreads();

    const int wave = tid >> 5;                 // 0..1
    const int lane = tid & 31;
    const int row_base = blockIdx.x * 32 + wave * 16;

    const int m  = lane & 15;                  // A row / D col index
    const int kh = (lane >> 4) * 2;            // lanes 0-15 -> K{0,1}, 16-31 -> K{2,3}
    const int kp = lane >> 4;                  // k-pair sub-offset

    v8f acc0 = {}, acc1 = {}, acc2 = {}, acc3 = {};
    const float* xrow = X + (size_t)(row_base + m) * K + kh;
    const float* wrow = sW + (m << 1);
    for (int k0 = 0; k0 < K; k0 += 4) {
        v2f a;
        a.x = xrow[k0];
        a.y = xrow[k0 + 1];
        const float* wp = wrow + (((k0 >> 1) + kp) << 7);
        v2f b0 = *(const v2f*)(wp + 0);
        v2f b1 = *(const v2f*)(wp + 32);
        v2f b2 = *(const v2f*)(wp + 64);
        v2f b3 = *(const v2f*)(wp + 96);
        acc0 = __builtin_amdgcn_wmma_f32_16x16x4_f32(false, a, false, b0, (short)0, acc0, false, false);
        acc1 = __builtin_amdgcn_wmma_f32_16x16x4_f32(false, a, false, b1, (short)0, acc1, false, false);
        acc2 = __builtin_amdgcn_wmma_f32_16x16x4_f32(false, a, false, b2, (short)0, acc2, false, false);
        acc3 = __builtin_amdgcn_wmma_f32_16x16x4_f32(false, a, false, b3, (short)0, acc3, false, false);
    }

    // D layout: VGPR r -> row r (lanes 0-15) / r+8 (lanes 16-31), col = lane&15
    const int rhalf = (lane >> 4) * 8;
    float* op = out + (size_t)row_base * HDIM + m;
#pragma unroll
    for (int r = 0; r < 8; ++r) {
        size_t ro = (size_t)(r + rhalf) * HDIM;
        op[ro +  0] = acc0[r];
        op[ro + 16] = acc1[r];
        op[ro + 32] = acc2[r];
        op[ro + 48] = acc3[r];
    }
}

// ---------------- edge scatter: agg[dst] += dinv[s]*dinv[d] * hw[src] ----------------
// 16 threads per edge, float4 per thread (global_load_b128 gather).
__global__ void scatter_edges(const float* __restrict__ hw, const float* __restrict__ dinv,
                              const int* __restrict__ src, const int* __restrict__ dst,
                              float* __restrict__ agg, int E) {
    long long tid = (long long)blockIdx.x * blockDim.x + threadIdx.x;
    long long total = (long long)E * 16;
    if (tid >= total) return;
    int e = (int)(tid >> 4);
    int j = (int)(tid & 15) * 4;
    int s = src[e];
    int d = dst[e];
    float nm = dinv[s] * dinv[d];
    const float4 v = *(const float4*)(hw + (size_t)s * HDIM + j);
    float* ap = agg + (size_t)d * HDIM + j;
    atomicAdd(ap + 0, nm * v.x);
    atomicAdd(ap + 1, nm * v.y);
    atomicAdd(ap + 2, nm * v.z);
    atomicAdd(ap + 3, nm * v.w);
}

// ---------------- self-loop + bias + relu (in place on agg) ----------------
__global__ void selfloop_bias_relu(float* __restrict__ agg, const float* __restrict__ hw,
                                   const float* __restrict__ dinv, const float* __restrict__ bias,
                                   int n) {
    long long tid = (long long)blockIdx.x * blockDim.x + threadIdx.x;
    long long total = (long long)n * 16;
    if (tid >= total) return;
    int i = (int)(tid >> 4);
    int j = (int)(tid & 15) * 4;
    float di = dinv[i];
    float d2 = di * di;
    size_t off = (size_t)i * HDIM + j;
    float4 a = *(const float4*)(agg + off);
    float4 h = *(const float4*)(hw + off);
    float4 b = *(const float4*)(bias + j);
    float4 r;
    r.x = a.x + d2 * h.x + b.x;
    r.y = a.y + d2 * h.y + b.y;
    r.z = a.z + d2 * h.z + b.z;
    r.w = a.w + d2 * h.w + b.w;
    r.x = r.x > 0.f ? r.x : 0.f;
    r.y = r.y > 0.f ? r.y : 0.f;
    r.z = r.z > 0.f ? r.z : 0.f;
    r.w = r.w > 0.f ? r.w : 0.f;
    *(float4*)(agg + off) = r;
}

// ---------------- pooling ----------------
__global__ void pool_kernel(const float* __restrict__ h, const int* __restrict__ batch,
                            float* __restrict__ pooled, float* __restrict__ cnt, int n) {
    long long tid = (long long)blockIdx.x * blockDim.x + threadIdx.x;
    long long total = (long long)n * 16;
    if (tid >= total) return;
    int i = (int)(tid >> 4);
    int j = (int)(tid & 15) * 4;
    int g = batch[i];
    const float4 v = *(const float4*)(h + (size_t)i * HDIM + j);
    float* pp = pooled + (size_t)g * HDIM + j;
    atomicAdd(pp + 0, v.x);
    atomicAdd(pp + 1, v.y);
    atomicAdd(pp + 2, v.z);
    atomicAdd(pp + 3, v.w);
    if ((tid & 15) == 0) atomicAdd(&cnt[g], 1.0f);
}

// ---------------- MLP head: [64,64] -> relu -> [64,2] ----------------
__global__ void __launch_bounds__(256)
head_kernel(const float* __restrict__ pooled, const float* __restrict__ cnt,
            const float* __restrict__ Wm1, const float* __restrict__ bm1,
            const float* __restrict__ Wm2, const float* __restrict__ bm2,
            float* __restrict__ out, int G) {
    __shared__ float avg[64 * HDIM];
    __shared__ float z[64 * HDIM];
    const int t = threadIdx.x;
    for (int idx = t; idx < G * HDIM; idx += 256) {
        int g = idx >> 6;
        float c = cnt[g];
        c = c < 1.0f ? 1.0f : c;
        avg[idx] = pooled[idx] / c;
    }
    __syncthreads();
    for (int idx = t; idx < G * HDIM; idx += 256) {
        int g = idx >> 6, h = idx & 63;
        float s = bm1[h];
#pragma unroll 8
        for (int k = 0; k < HDIM; ++k) s += avg[g * HDIM + k] * Wm1[k * HDIM + h];
        z[idx] = s > 0.0f ? s : 0.0f;
    }
    __syncthreads();
    if (t < G * 2) {
        int g = t >> 1, o = t & 1;
        float s = bm2[o];
#pragma unroll 8
        for (int k = 0; k < HDIM; ++k) s += z[g * HDIM + k] * Wm2[k * 2 + o];
        out[g * 2 + o] = s;
    }
}

extern "C" void kernel_launch(void* const* d_in, const int* in_sizes, int n_in,
                              void* d_out, int out_size, void* d_ws, size_t ws_size,
                              hipStream_t stream) {
    const float* x   = (const float*)d_in[0];
    const int*   ei  = (const int*)d_in[1];   // int64 in source, JAX x64 off -> int32
    const int*   bat = (const int*)d_in[2];
    const float* W1  = (const float*)d_in[3];
    const float* b1  = (const float*)d_in[4];
    const float* W2  = (const float*)d_in[5];
    const float* b2  = (const float*)d_in[6];
    const float* W3  = (const float*)d_in[7];
    const float* b3  = (const float*)d_in[8];
    const float* Wm1 = (const float*)d_in[9];
    const float* bm1 = (const float*)d_in[10];
    const float* Wm2 = (const float*)d_in[11];
    const float* bm2 = (const float*)d_in[12];
    float* out = (float*)d_out;

    const int N  = in_sizes[2];            // 100000 (divisible by 32)
    const int E  = in_sizes[1] / 2;        // 1600000
    const int IN = in_sizes[0] / N;        // 128
    const int G  = out_size / 2;           // 64

    const int* src = ei;
    const int* dst = ei + E;

    // workspace layout (floats), offsets 256B-aligned
    float* ws     = (float*)d_ws;
    float* dinv   = ws;                        // N
    float* bufA   = ws + 102400;               // N*64  (GEMM output)
    float* bufB   = bufA + 6400000;            // N*64  (aggregation / layer output)
    float* pooled = bufB + 6400000;            // G*64
    float* cnt    = pooled + 4096;             // G

    const long long NH   = (long long)N * HDIM;
    const long long N16  = (long long)N * 16;   // float4 granularity over N*64
    const long long E16  = (long long)E * 16;
    const int B256 = 256;
    const int gN    = (N + B256 - 1) / B256;
    const int gE    = (E + B256 - 1) / B256;
    const int gN16  = (int)((N16 + B256 - 1) / B256);
    const int gE16  = (int)((E16 + B256 - 1) / B256);
    const int gZero = (int)((NH / 4 + B256 - 1) / B256);
    const int gGemm = N / 32;

    // degrees -> dinv
    zero_f32v4<<<(N / 4 + B256 - 1) / B256, B256, 0, stream>>>((float4*)dinv, N / 4);
    deg_kernel<<<gE, B256, 0, stream>>>(dst, dinv, E);
    dinv_kernel<<<gN, B256, 0, stream>>>(dinv, N);

    // ---- layer 1 ----
    gemm_wmma_f32<<<gGemm, 64, 0, stream>>>(x, W1, bufA, IN);
    zero_f32v4<<<gZero, B256, 0, stream>>>((float4*)bufB, NH / 4);
    scatter_edges<<<gE16, B256, 0, stream>>>(bufA, dinv, src, dst, bufB, E);
    selfloop_bias_relu<<<gN16, B256, 0, stream>>>(bufB, bufA, dinv, b1, N);

    // ---- layer 2 ----
    gemm_wmma_f32<<<gGemm, 64, 0, stream>>>(bufB, W2, bufA, HDIM);
    zero_f32v4<<<gZero, B256, 0, stream>>>((float4*)bufB, NH / 4);
    scatter_edges<<<gE16, B256, 0, stream>>>(bufA, dinv, src, dst, bufB, E);
    selfloop_bias_relu<<<gN16, B256, 0, stream>>>(bufB, bufA, dinv, b2, N);

    // ---- layer 3 ----
    gemm_wmma_f32<<<gGemm, 64, 0, stream>>>(bufB, W3, bufA, HDIM);
    zero_f32v4<<<gZero, B256, 0, stream>>>((float4*)bufB, NH / 4);
    scatter_edges<<<gE16, B256, 0, stream>>>(bufA, dinv, src, dst, bufB, E);
    selfloop_bias_relu<<<gN16, B256, 0, stream>>>(bufB, bufA, dinv, b3, N);

    // ---- pooling + head ----
    zero_f32v4<<<((G * HDIM + G) / 4 + B256 - 1) / B256, B256, 0, stream>>>(
        (float4*)pooled, (G * HDIM + G) / 4);
    pool_kernel<<<gN16, B256, 0, stream>>>(bufB, bat, pooled, cnt, N);
    head_kernel<<<1, B256, 0, stream>>>(pooled, cnt, Wm1, bm1, Wm2, bm2, out, G);
}